// HMM_bayes_25537875542067
// MI455X (gfx1250) — compile-verified
//
#include <hip/hip_runtime.h>
#include <math.h>

#define T_TOT 65536
#define S_TOT 512
#define M_DIM 4

typedef float v2f __attribute__((ext_vector_type(2)));
typedef float v8f __attribute__((ext_vector_type(8)));

// ---- optional CDNA5 async global->LDS path (guarded; falls back cleanly) ----
#if defined(__HIP_DEVICE_COMPILE__)
#if __has_builtin(__builtin_amdgcn_global_load_async_to_lds_b32)
#define ASYNC_E 1
#endif
#endif
#ifndef ASYNC_E
#define ASYNC_E 0
#endif

#if ASYNC_E
// builtin expects (addrspace(1) int*, addrspace(3) int*, imm offset, imm cpol)
#define GAS(p) ((__attribute__((address_space(1))) int*)(p))
#define LAS(p) ((__attribute__((address_space(3))) int*)(p))
#if __has_builtin(__builtin_amdgcn_s_wait_asynccnt)
#define WAIT_ASYNC(n) __builtin_amdgcn_s_wait_asynccnt(n)
#else
#define WAIT_ASYNC(n) asm volatile("s_wait_asynccnt %0" ::"n"(n))
#endif
#endif

// ---------------------------------------------------------------------------
// Kernel A: emission precompute with V_WMMA_F32_16X16X4_F32 (K=4 == M_DIM).
// Per workgroup: 16 timesteps x 512 states; 32 waves, one 16x16 WMMA tile
// each, -sum(lam) bias baked into the C accumulator. Then per-t row max c_t,
// store E[t,s] = exp(e - c_t) and cprime[t] = c_t - sum_m lgamma(x[t,m]+1).
// ---------------------------------------------------------------------------
__global__ __launch_bounds__(1024) void hmm_emit_kernel(
    const int* __restrict__ x,
    const float* __restrict__ lam0, const float* __restrict__ lam1,
    const float* __restrict__ lam2,
    float* __restrict__ E, float* __restrict__ cprime)
{
    __shared__ float rmax[32][16];
    __shared__ float cmax[16];

    const int tid  = threadIdx.x;
    const int wave = tid >> 5;
    const int lane = tid & 31;
    const int half = lane >> 4;      // 0: K={0,1}/rows 0..7 ; 1: K={2,3}/rows 8..15
    const int l15  = lane & 15;
    const int t0   = blockIdx.x * 16;
    const int n    = wave * 16 + l15;

    const int s0 = n >> 6, s1 = (n >> 3) & 7, s2 = n & 7;
    float llog[4], lsum = 0.f;
#pragma unroll
    for (int m = 0; m < 4; ++m) {
        float lc = lam0[s0 * 4 + m] + lam1[s1 * 4 + m] + lam2[s2 * 4 + m];
        lsum += lc;
        llog[m] = __logf(lc);
    }

    const int kb = half * 2;
    v2f a, b;
    a.x = (float)x[(t0 + l15) * M_DIM + kb];
    a.y = (float)x[(t0 + l15) * M_DIM + kb + 1];
    b.x = llog[kb];
    b.y = llog[kb + 1];
    v8f c;
#pragma unroll
    for (int r = 0; r < 8; ++r) c[r] = -lsum;

    v8f d = __builtin_amdgcn_wmma_f32_16x16x4_f32(
        false, a, false, b, (short)0, c, false, false);

    // row max over this wave's 16 columns (masks<16: halves reduce separately)
    float m8[8];
#pragma unroll
    for (int r = 0; r < 8; ++r) {
        float v = d[r];
        v = fmaxf(v, __shfl_xor(v, 1));
        v = fmaxf(v, __shfl_xor(v, 2));
        v = fmaxf(v, __shfl_xor(v, 4));
        v = fmaxf(v, __shfl_xor(v, 8));
        m8[r] = v;
    }
    if (l15 == 0) {
#pragma unroll
        for (int r = 0; r < 8; ++r) rmax[wave][r + 8 * half] = m8[r];
    }
    __syncthreads();

    if (tid < 16) {
        float v = rmax[0][tid];
        for (int w = 1; w < 32; ++w) v = fmaxf(v, rmax[w][tid]);
        cmax[tid] = v;
        float lg = 0.f;
#pragma unroll
        for (int m = 0; m < 4; ++m)
            lg += lgammaf((float)x[(t0 + tid) * M_DIM + m] + 1.f);
        cprime[t0 + tid] = v - lg;
    }
    __syncthreads();

#pragma unroll
    for (int r = 0; r < 8; ++r) {
        int row = r + 8 * half;
        E[(size_t)(t0 + row) * S_TOT + n] = __expf(d[r] - cmax[row]);
    }
}

// ---------------------------------------------------------------------------
// Kernel B: persistent single-workgroup forward scan (512 threads, 16 waves).
// Probability-space recurrence with per-step rescaling; Kronecker-factorized
// transition (P0 (x) P1 (x) P2) applied as three 8-wide mode products in LDS.
// 4 barriers/step; E rows software-pipelined (async global->LDS when present).
// ---------------------------------------------------------------------------
__global__ __launch_bounds__(512) void hmm_scan_kernel(
    const int* __restrict__ x,
    const float* __restrict__ lam0, const float* __restrict__ lam1,
    const float* __restrict__ lam2,
    const float* __restrict__ logA0, const float* __restrict__ logA1,
    const float* __restrict__ logA2,
    const float* __restrict__ logp0, const float* __restrict__ logp1,
    const float* __restrict__ logp2,
    const float* __restrict__ E, const float* __restrict__ cpr,
    float* __restrict__ out, int use_pre)
{
    __shared__ float P0[64], P1[64], P2[64];
    __shared__ float pr0[8], pr1[8], pr2[8];
    __shared__ float bufA[S_TOT], bufB[S_TOT];
    __shared__ float redsum[16], redmax[16];
    __shared__ float lgtab[32];
#if ASYNC_E
    __shared__ float Ebuf[2][S_TOT];
#endif

    const int tid  = threadIdx.x;
    const int wave = tid >> 5;
    const int lane = tid & 31;
    const int j0 = tid >> 6, j1 = (tid >> 3) & 7, j2 = tid & 7;

    // ---- one-time init: column softmax (axis 0) of logA, priors, lgamma table
    if (tid < 24) {
        const float* A = (tid < 8) ? logA0 : (tid < 16) ? logA1 : logA2;
        float*       P = (tid < 8) ? P0    : (tid < 16) ? P1    : P2;
        int k = tid & 7;
        float mx = -3.4e38f;
        for (int j = 0; j < 8; ++j) mx = fmaxf(mx, A[j * 8 + k]);
        float sum = 0.f;
        for (int j = 0; j < 8; ++j) sum += __expf(A[j * 8 + k] - mx);
        float inv = 1.f / sum;
        for (int j = 0; j < 8; ++j) P[j * 8 + k] = __expf(A[j * 8 + k] - mx) * inv;
    } else if (tid < 27) {
        const float* p = (tid == 24) ? logp0 : (tid == 25) ? logp1 : logp2;
        float*       o = (tid == 24) ? pr0   : (tid == 25) ? pr1   : pr2;
        float mx = -3.4e38f;
        for (int j = 0; j < 8; ++j) mx = fmaxf(mx, p[j]);
        float s = 0.f;
        for (int j = 0; j < 8; ++j) s += __expf(p[j] - mx);
        float inv = 1.f / s;
        for (int j = 0; j < 8; ++j) o[j] = __expf(p[j] - mx) * inv;
    } else if (tid >= 32 && tid < 64) {
        lgtab[tid - 32] = lgammaf((float)(tid - 32) + 1.f);
    }
    float llog[4], lsum = 0.f;    // inline-emission fallback params
#pragma unroll
    for (int m = 0; m < 4; ++m) {
        float lc = lam0[j0 * 4 + m] + lam1[j1 * 4 + m] + lam2[j2 * 4 + m];
        lsum += lc;
        llog[m] = __logf(lc);
    }
    __syncthreads();

    double logZ = 0.0;               // accumulated on tid 0 only (f64)

    // ---- t = 0 : u = prior * exp(e0 - c0)
    {
        float u, cadd;
        float prior = pr0[j0] * pr1[j1] * pr2[j2];
        if (use_pre) {
            u = prior * E[tid];
            cadd = cpr[0];
        } else {
            float e = -lsum;
            int xt[4];
#pragma unroll
            for (int m = 0; m < 4; ++m) { xt[m] = x[m]; e += (float)xt[m] * llog[m]; }
            float v = e;
            v = fmaxf(v, __shfl_xor(v, 16)); v = fmaxf(v, __shfl_xor(v, 8));
            v = fmaxf(v, __shfl_xor(v, 4));  v = fmaxf(v, __shfl_xor(v, 2));
            v = fmaxf(v, __shfl_xor(v, 1));
            if (lane == 0) redmax[wave] = v;
            __syncthreads();
            float r = redmax[lane & 15];
            r = fmaxf(r, __shfl_xor(r, 8)); r = fmaxf(r, __shfl_xor(r, 4));
            r = fmaxf(r, __shfl_xor(r, 2)); r = fmaxf(r, __shfl_xor(r, 1));
            u = prior * __expf(e - r);
            cadd = r - (lgtab[xt[0]] + lgtab[xt[1]] + lgtab[xt[2]] + lgtab[xt[3]]);
        }
        float v = u;
        v += __shfl_xor(v, 16); v += __shfl_xor(v, 8); v += __shfl_xor(v, 4);
        v += __shfl_xor(v, 2);  v += __shfl_xor(v, 1);
        if (lane == 0) redsum[wave] = v;
        __syncthreads();
        float r = redsum[lane & 15];
        r += __shfl_xor(r, 8); r += __shfl_xor(r, 4);
        r += __shfl_xor(r, 2); r += __shfl_xor(r, 1);
        float Z = r;
        if (tid == 0) logZ = (double)cadd + (double)__logf(Z);
        bufA[tid] = u / Z;
        __syncthreads();
    }

    // ---- prime the E-row pipeline (row 1)
    float Ecur = 0.f, Enext = 0.f, ccur = 0.f, cnext = 0.f;
    if (use_pre) {
#if ASYNC_E
        __builtin_amdgcn_global_load_async_to_lds_b32(
            GAS(&E[S_TOT + tid]), LAS(&Ebuf[1][tid]), 0, 0);
#else
        Enext = E[S_TOT + tid];
#endif
        cnext = cpr[1];
    }

    // ---- main scan over t = 1..T-1
    for (int t = 1; t < T_TOT; ++t) {
        if (use_pre) {                       // kick fetch of row t+1
            ccur = cnext;
            if (t + 1 < T_TOT) {
#if ASYNC_E
                __builtin_amdgcn_global_load_async_to_lds_b32(
                    GAS(&E[(size_t)(t + 1) * S_TOT + tid]),
                    LAS(&Ebuf[(t + 1) & 1][tid]), 0, 0);
#else
                Ecur  = Enext;
                Enext = E[(size_t)(t + 1) * S_TOT + tid];
                if (t + 16 < T_TOT)
                    __builtin_prefetch(&E[(size_t)(t + 16) * S_TOT + tid], 0, 0);
#endif
                cnext = cpr[t + 1];
            }
#if !ASYNC_E
            else { Ecur = Enext; }
#endif
        }

        // stage 1: contract k0 (stride-64 in LDS)
        {
            int rem = tid & 63;
            float acc = 0.f;
#pragma unroll
            for (int k = 0; k < 8; ++k) acc += P0[j0 * 8 + k] * bufA[k * 64 + rem];
            bufB[tid] = acc;
        }
        __syncthreads();
        // stage 2: contract k1 (stride-8)
        {
            int base = (tid & ~63) | (tid & 7);
            float acc = 0.f;
#pragma unroll
            for (int k = 0; k < 8; ++k) acc += P1[j1 * 8 + k] * bufB[base + k * 8];
            bufA[tid] = acc;
        }
        __syncthreads();
        // stage 3: contract k2 (stride-1)
        float w;
        {
            int base = tid & ~7;
            float acc = 0.f;
#pragma unroll
            for (int k = 0; k < 8; ++k) acc += P2[j2 * 8 + k] * bufA[base + k];
            w = acc;
        }

        float u, cadd;
        if (use_pre) {
#if ASYNC_E
            if (t + 1 < T_TOT) { WAIT_ASYNC(1); } else { WAIT_ASYNC(0); }
            Ecur = Ebuf[t & 1][tid];
#endif
            u = w * Ecur;
            cadd = ccur;
        } else {
            float e = -lsum;
            int xt[4];
#pragma unroll
            for (int m = 0; m < 4; ++m) { xt[m] = x[t * M_DIM + m]; e += (float)xt[m] * llog[m]; }
            float v = e;
            v = fmaxf(v, __shfl_xor(v, 16)); v = fmaxf(v, __shfl_xor(v, 8));
            v = fmaxf(v, __shfl_xor(v, 4));  v = fmaxf(v, __shfl_xor(v, 2));
            v = fmaxf(v, __shfl_xor(v, 1));
            if (lane == 0) redmax[wave] = v;
            __syncthreads();
            float r = redmax[lane & 15];
            r = fmaxf(r, __shfl_xor(r, 8)); r = fmaxf(r, __shfl_xor(r, 4));
            r = fmaxf(r, __shfl_xor(r, 2)); r = fmaxf(r, __shfl_xor(r, 1));
            u = w * __expf(e - r);
            cadd = r - (lgtab[xt[0]] + lgtab[xt[1]] + lgtab[xt[2]] + lgtab[xt[3]]);
        }

        // Z = sum_s u (wave partials -> all-wave redundant combine: no bcast barrier)
        float v = u;
        v += __shfl_xor(v, 16); v += __shfl_xor(v, 8); v += __shfl_xor(v, 4);
        v += __shfl_xor(v, 2);  v += __shfl_xor(v, 1);
        if (lane == 0) redsum[wave] = v;
        __syncthreads();
        float r = redsum[lane & 15];
        r += __shfl_xor(r, 8); r += __shfl_xor(r, 4);
        r += __shfl_xor(r, 2); r += __shfl_xor(r, 1);
        float Z = r;
        if (tid == 0) logZ += (double)cadd + (double)__logf(Z);
        bufA[tid] = u / Z;
        __syncthreads();
    }

    if (tid == 0) out[0] = (float)logZ;
}

// ---------------------------------------------------------------------------
extern "C" void kernel_launch(void* const* d_in, const int* in_sizes, int n_in,
                              void* d_out, int out_size, void* d_ws, size_t ws_size,
                              hipStream_t stream) {
    (void)in_sizes; (void)n_in; (void)out_size;
    const int*   x    = (const int*)d_in[0];
    const float* lam0 = (const float*)d_in[1];
    const float* lam1 = (const float*)d_in[2];
    const float* lam2 = (const float*)d_in[3];
    const float* A0   = (const float*)d_in[4];
    const float* A1   = (const float*)d_in[5];
    const float* A2   = (const float*)d_in[6];
    const float* p0   = (const float*)d_in[7];
    const float* p1   = (const float*)d_in[8];
    const float* p2   = (const float*)d_in[9];
    float* out = (float*)d_out;

    const size_t needE = (size_t)T_TOT * S_TOT * sizeof(float);   // 128 MB (L2-resident)
    const size_t needC = (size_t)T_TOT * sizeof(float);
    const int use_pre = (ws_size >= needE + needC) ? 1 : 0;
    float* E   = (float*)d_ws;
    float* cpr = (float*)((char*)d_ws + needE);

    if (use_pre) {
        hipLaunchKernelGGL(hmm_emit_kernel, dim3(T_TOT / 16), dim3(1024), 0, stream,
                           x, lam0, lam1, lam2, E, cpr);
    }
    hipLaunchKernelGGL(hmm_scan_kernel, dim3(1), dim3(512), 0, stream,
                       x, lam0, lam1, lam2, A0, A1, A2, p0, p1, p2,
                       E, cpr, out, use_pre);
}